// MultiHeadAttention_26482768347194
// MI455X (gfx1250) — compile-verified
//
#include <hip/hip_runtime.h>

// ---------------------------------------------------------------------------
// MultiHeadAttention (buggy softmax over size-1 axis => score == 1 exactly).
// Exact algebraic rewrite:
//   xs[b,:]  = sum_l x[b,l,:]                       (stream x, 16 MB)
//   vs[b,:]  = xs[b,:] @ wv + L*bv                  (16x512 @ 512x4096 WMMA)
//   o2[b,:]  = vs[b,:] @ fc_w + fc_b                (16x4096 @ 4096x512 WMMA)
//   out[b,l] = LN(o2[b,:] + x[b,l,:]) * g + beta    (stream x + out, 32 MB)
// ~64 MB total traffic => ~2.7 us at 23.3 TB/s; GEMMs are ~67 MFLOP (noise).
// q/k/score_w/score_b paths are dead code (score==1 regardless of tanh).
// ---------------------------------------------------------------------------

typedef __attribute__((ext_vector_type(2))) float v2f;
typedef __attribute__((ext_vector_type(8))) float v8f;

#define B_     8
#define L_     1024
#define D_     512
#define HD_    4096
#define NCHUNK 16
#define LCH    (L_ / NCHUNK) // 64

// ---- pass 1a: per-(batch, L-chunk) column partial sums of x ----------------
__global__ __launch_bounds__(512) void k_colsum_partial(
    const float* __restrict__ x, float* __restrict__ partial) {
  const int chunk = blockIdx.x & (NCHUNK - 1);
  const int b     = blockIdx.x >> 4;
  const int col   = threadIdx.x; // 0..511
  const float* xp = x + ((size_t)b * L_ + (size_t)chunk * LCH) * D_ + col;
  float s = 0.0f;
#pragma unroll 4
  for (int l = 0; l < LCH; ++l) s += xp[(size_t)l * D_];
  partial[((size_t)chunk * B_ + b) * D_ + col] = s;
}

// ---- pass 1b: reduce chunks -> xs16 (16x512, rows 8..15 zero-padded) -------
__global__ __launch_bounds__(256) void k_colsum_reduce(
    const float* __restrict__ partial, float* __restrict__ xs16) {
  const int idx = blockIdx.x * 256 + threadIdx.x; // 0..8191
  const int m   = idx >> 9;
  const int col = idx & (D_ - 1);
  float s = 0.0f;
  if (m < B_) {
#pragma unroll
    for (int c = 0; c < NCHUNK; ++c)
      s += partial[((size_t)c * B_ + m) * D_ + col];
  }
  xs16[idx] = s; // rows 8..15 = 0 (WMMA pad)
}

// ---- GEMM1: vs16(16x4096) = xs16(16x512) @ wv(512x4096) + L*bv -------------
// One wave per 16-col tile. V_WMMA_F32_16X16X4_F32, K stepped by 4.
__global__ __launch_bounds__(32) void k_gemm1(
    const float* __restrict__ xs16, const float* __restrict__ wv,
    const float* __restrict__ bv, float* __restrict__ vs16) {
  const int lane = threadIdx.x;
  const int m    = lane & 15;
  const int g    = lane >> 4;                  // K half-group
  const int n    = blockIdx.x * 16 + m;        // output column (lanes 0..15 pattern)
  const float* arow = xs16 + (size_t)m * D_;
  v8f c = {};
  for (int k = 0; k < D_; k += 4) {
    const int kA = k + 2 * g;                  // this half-wave's K pair
    v2f a, bb;
    a.x  = arow[kA];
    a.y  = arow[kA + 1];
    bb.x = wv[(size_t)kA * HD_ + n];
    bb.y = wv[(size_t)(kA + 1) * HD_ + n];
    c = __builtin_amdgcn_wmma_f32_16x16x4_f32(
        false, a, false, bb, (short)0, c, false, false);
  }
  const float bias = (float)L_ * bv[n];
#pragma unroll
  for (int r = 0; r < 8; ++r)                  // VGPR r -> row r + 8*g
    vs16[(size_t)(r + 8 * g) * HD_ + n] = c[r] + bias;
}

// ---- GEMM2: o2(16x512) = vs16(16x4096) @ fc_w(4096x512) + fc_b -------------
__global__ __launch_bounds__(32) void k_gemm2(
    const float* __restrict__ vs16, const float* __restrict__ fc_w,
    const float* __restrict__ fc_b, float* __restrict__ o2) {
  const int lane = threadIdx.x;
  const int m    = lane & 15;
  const int g    = lane >> 4;
  const int n    = blockIdx.x * 16 + m;
  const float* arow = vs16 + (size_t)m * HD_;
  v8f c = {};
  for (int k = 0; k < HD_; k += 4) {
    const int kA = k + 2 * g;
    v2f a, bb;
    a.x  = arow[kA];
    a.y  = arow[kA + 1];
    bb.x = fc_w[(size_t)kA * D_ + n];
    bb.y = fc_w[(size_t)(kA + 1) * D_ + n];
    c = __builtin_amdgcn_wmma_f32_16x16x4_f32(
        false, a, false, bb, (short)0, c, false, false);
  }
  const float bias = fc_b[n];
#pragma unroll
  for (int r = 0; r < 8; ++r)
    o2[(size_t)(r + 8 * g) * D_ + n] = c[r] + bias;
}

// ---- pass 2: out[b,l,:] = LN(o2[b,:] + x[b,l,:]) * g + beta ----------------
__global__ __launch_bounds__(128) void k_residual_ln(
    const float* __restrict__ x, const float* __restrict__ o2,
    const float* __restrict__ g, const float* __restrict__ beta,
    float* __restrict__ out) {
  __shared__ float ssum[4];
  __shared__ float ssq[4];
  const int row = blockIdx.x;  // b*1024 + l
  const int b   = row >> 10;
  const int t   = threadIdx.x; // 0..127, float4 per thread
  const float4 xv = ((const float4*)(x  + (size_t)row * D_))[t];
  const float4 ov = ((const float4*)(o2 + (size_t)b   * D_))[t];
  float4 y;
  y.x = xv.x + ov.x; y.y = xv.y + ov.y;
  y.z = xv.z + ov.z; y.w = xv.w + ov.w;
  float s = y.x + y.y + y.z + y.w;
  float q = y.x * y.x + y.y * y.y + y.z * y.z + y.w * y.w;
#pragma unroll
  for (int off = 16; off > 0; off >>= 1) { // wave32 butterfly
    s += __shfl_xor(s, off, 32);
    q += __shfl_xor(q, off, 32);
  }
  const int w = t >> 5;
  if ((t & 31) == 0) { ssum[w] = s; ssq[w] = q; }
  __syncthreads();
  s = ssum[0] + ssum[1] + ssum[2] + ssum[3];
  q = ssq[0]  + ssq[1]  + ssq[2]  + ssq[3];
  const float inv  = 1.0f / (float)D_;
  const float mu   = s * inv;
  const float var  = fmaxf(q * inv - mu * mu, 0.0f);
  const float rstd = rsqrtf(var + 1e-5f);
  const float4 gv = ((const float4*)g)[t];
  const float4 bv = ((const float4*)beta)[t];
  float4 o;
  o.x = (y.x - mu) * rstd * gv.x + bv.x;
  o.y = (y.y - mu) * rstd * gv.y + bv.y;
  o.z = (y.z - mu) * rstd * gv.z + bv.z;
  o.w = (y.w - mu) * rstd * gv.w + bv.w;
  ((float4*)(out + (size_t)row * D_))[t] = o;
}

extern "C" void kernel_launch(void* const* d_in, const int* in_sizes, int n_in,
                              void* d_out, int out_size, void* d_ws, size_t ws_size,
                              hipStream_t stream) {
  (void)in_sizes; (void)n_in; (void)out_size; (void)ws_size;
  // setup_inputs order: 0 input, 1 wq, 2 bq, 3 wk, 4 bk, 5 wv, 6 bv,
  //                     7 score_w, 8 score_b, 9 fc_w, 10 fc_b, 11 ln_g, 12 ln_b
  const float* x    = (const float*)d_in[0];
  const float* wv   = (const float*)d_in[5];
  const float* bv   = (const float*)d_in[6];
  const float* fc_w = (const float*)d_in[9];
  const float* fc_b = (const float*)d_in[10];
  const float* ln_g = (const float*)d_in[11];
  const float* ln_b = (const float*)d_in[12];
  float* out = (float*)d_out;

  // workspace layout (floats): partial[16*8*512] | xs16[16*512] | vs16[16*4096] | o2[16*512]
  float* ws      = (float*)d_ws;
  float* partial = ws;                               // 65536 floats
  float* xs16    = partial + (size_t)NCHUNK * B_ * D_; // 8192 floats
  float* vs16    = xs16 + 16 * D_;                   // 65536 floats
  float* o2      = vs16 + 16 * HD_;                  // 8192 floats

  k_colsum_partial<<<B_ * NCHUNK, 512, 0, stream>>>(x, partial);
  k_colsum_reduce <<<32, 256, 0, stream>>>(partial, xs16);
  k_gemm1         <<<HD_ / 16, 32, 0, stream>>>(xs16, wv, bv, vs16);
  k_gemm2         <<<D_ / 16, 32, 0, stream>>>(vs16, fc_w, fc_b, o2);
  k_residual_ln   <<<B_ * L_, 128, 0, stream>>>(x, o2, ln_g, ln_b, out);
}